// BiaffineLinker_14199161881073
// MI455X (gfx1250) — compile-verified
//
#include <hip/hip_runtime.h>
#include <hip/hip_bf16.h>
#include <math.h>

typedef __attribute__((ext_vector_type(2))) float v2f;
typedef __attribute__((ext_vector_type(8))) float v8f;
typedef __attribute__((ext_vector_type(4))) int   v4i;

#define EPS 1e-8f
#define B_  4
#define S_  2048
#define H_  1024
#define K_  128

// ---------------------------------------------------------------------------
// CDNA5 async global->LDS staging (ASYNCcnt-tracked), with safe fallback.
// Signature (from hipcc diagnostic): param0 = v4i __device__* (AS1, non-const),
// param1 = v4i __shared__* (AS3), then imm offset, imm cpol.
// ---------------------------------------------------------------------------
#if defined(__has_builtin)
#  if __has_builtin(__builtin_amdgcn_global_load_async_to_lds_b128)
#    define USE_ASYNC_LDS 1
#  endif
#endif
#ifndef USE_ASYNC_LDS
#  define USE_ASYNC_LDS 0
#endif

#if USE_ASYNC_LDS
typedef __attribute__((address_space(1))) v4i gv4i;
typedef __attribute__((address_space(3))) v4i lv4i;

__device__ __forceinline__ void async_copy16(void* lds_dst, const void* gsrc) {
    // LDS generic pointers carry the LDS byte offset in the low 32 bits
    // -> value-truncating cast.
    lv4i* ldst = (lv4i*)(unsigned int)(unsigned long long)lds_dst;
    gv4i* gptr = (gv4i*)(unsigned long long)gsrc;   // drops const via int round-trip
    __builtin_amdgcn_global_load_async_to_lds_b128(gptr, ldst, 0, 0);
}
#endif

__device__ __forceinline__ void wait_async0() {
#if USE_ASYNC_LDS
#  if __has_builtin(__builtin_amdgcn_s_wait_asynccnt)
    __builtin_amdgcn_s_wait_asynccnt(0);
#  else
    asm volatile("s_wait_asynccnt 0" ::: "memory");
#  endif
#endif
}

// ---------------------------------------------------------------------------
// WMMA helper: D = A(16x4 f32) x B(4x16 f32) + C(16x16 f32)
// ---------------------------------------------------------------------------
__device__ __forceinline__ v8f wmma_f32(v2f a, v2f b, v8f c) {
    return __builtin_amdgcn_wmma_f32_16x16x4_f32(false, a, false, b,
                                                 (short)0, c, false, false);
}

__device__ __forceinline__ unsigned int ord_f32(float f) {
    unsigned int u = __float_as_uint(f);
    return (u & 0x80000000u) ? ~u : (u | 0x80000000u);
}

// ---------------------------------------------------------------------------
// Kernel 1: per (batch, class) top-K=128 selection.
// Packed sort key: (ordered_conf << 32) | (0xFFFFFFFF - token_idx)
// -> descending conf, ties broken by lower index (matches jax.lax.top_k).
// ---------------------------------------------------------------------------
__global__ __launch_bounds__(256)
void topk_kernel(const float* __restrict__ logits, const int* __restrict__ amask,
                 int* __restrict__ rowsKey, int* __restrict__ rowsVal,
                 float* __restrict__ out)
{
    __shared__ unsigned long long keys[S_];
    __shared__ unsigned long long red[256];
    const int t    = threadIdx.x;
    const int b    = blockIdx.x >> 1;
    const int type = blockIdx.x & 1;          // 0 => keys (class 1), 1 => values (class 2)
    const int cls  = type + 1;

    for (int s = t; s < S_; s += 256) {
        const float* lp = logits + ((size_t)b * S_ + s) * 3;
        float l0 = lp[0], l1 = lp[1], l2 = lp[2];
        int pred = 0; float best = l0;
        if (l1 > best) { pred = 1; best = l1; }
        if (l2 > best) { pred = 2; best = l2; }
        float conf;
        if (pred == cls && amask[b * S_ + s] == 1) {
            float e0 = expf(l0 - best), e1 = expf(l1 - best), e2 = expf(l2 - best);
            conf = (cls == 1 ? e1 : e2) / (e0 + e1 + e2);
        } else {
            conf = -__builtin_huge_valf();
        }
        keys[s] = ((unsigned long long)ord_f32(conf) << 32) |
                  (unsigned long long)(0xFFFFFFFFu - (unsigned)s);
    }
    __syncthreads();

    int* rows = type ? rowsVal : rowsKey;
    int* outI = (int*)out + B_ * K_ * K_ + type * (B_ * K_) + b * K_;

    for (int i = 0; i < K_; ++i) {
        unsigned long long local = 0ull;
        for (int s = t; s < S_; s += 256)
            if (keys[s] > local) local = keys[s];
        red[t] = local;
        __syncthreads();
        for (int off = 128; off > 0; off >>= 1) {
            if (t < off) { if (red[t + off] > red[t]) red[t] = red[t + off]; }
            __syncthreads();
        }
        unsigned long long win = red[0];
        int idx = (int)(0xFFFFFFFFu - (unsigned)(win & 0xFFFFFFFFull));
        if (t == 0) {
            outI[i]         = idx;
            rows[b*K_ + i]  = b * S_ + idx;   // flat token index into [B*S, ...]
            keys[idx]       = 0ull;           // 0 < any live key
        }
        __syncthreads();
    }
}

// ---------------------------------------------------------------------------
// Kernel 2: C[M,N] = gather_rows(Abase, rowIdx) @ W[Kd,N] (+ bias)
// Block tile 128x64, 8 waves, wave tile 16x64. Double-buffered LDS:
//   A tile staged via GLOBAL_LOAD_ASYNC_TO_LDS_B128 (ASYNCcnt),
//   B tile stored K-pair-interleaved so each WMMA B-fragment is ONE
//   contiguous 8-byte ds_load into an even-aligned VGPR pair.
// ---------------------------------------------------------------------------
#define ASTRIDE 20   // 80B row stride: 16B-aligned b128 targets, 16 distinct banks

__global__ __launch_bounds__(256)
void gemm_rows_wmma(const float* __restrict__ Abase, const int* __restrict__ rowIdx,
                    const float* __restrict__ W, const float* __restrict__ bias,
                    float* __restrict__ C, int M, int N, int Kd)
{
    __shared__ float As[2][128][ASTRIDE];
    __shared__ float Bs[2][8][132];      // (k,c) -> Bs[k>>1][c*2 + (k&1)]
    __shared__ int   rows[128];

    const int t     = threadIdx.x;
    const int lane  = t & 31;
    const int wave  = t >> 5;
    const int mBase = blockIdx.y * 128;
    const int nBase = blockIdx.x * 64;

    if (t < 128) {
        int r = mBase + t;
        rows[t] = rowIdx ? rowIdx[r] : r;
    }
    __syncthreads();

    v8f acc0 = {}, acc1 = {}, acc2 = {}, acc3 = {};

    const int ml   = lane & 15;
    const int hf   = lane >> 4;
    const int mRow = wave * 16;

    const int arow = t >> 1;             // A staging: 128 rows x 16 cols
    const int acol = (t & 1) * 8;
    const int brow = t >> 4;             // B staging: 16 rows x 64 cols
    const int bcol = (t & 15) * 4;
    const int bp   = brow >> 1;          // K-pair index
    const int bo   = brow & 1;

    // ---- tile stage: async A (global->LDS DMA), interleaved B ----
    auto stage = [&](int buf, int k0) {
        const float* asrc = Abase + (size_t)rows[arow] * Kd + k0 + acol;
#if USE_ASYNC_LDS
        async_copy16(&As[buf][arow][acol + 0], asrc + 0);
        async_copy16(&As[buf][arow][acol + 4], asrc + 4);
#else
        float4 f0 = *(const float4*)(asrc);
        float4 f1 = *(const float4*)(asrc + 4);
        As[buf][arow][acol+0] = f0.x; As[buf][arow][acol+1] = f0.y;
        As[buf][arow][acol+2] = f0.z; As[buf][arow][acol+3] = f0.w;
        As[buf][arow][acol+4] = f1.x; As[buf][arow][acol+5] = f1.y;
        As[buf][arow][acol+6] = f1.z; As[buf][arow][acol+7] = f1.w;
#endif
        const float* bsrc = W + (size_t)(k0 + brow) * N + nBase + bcol;
        float4 f = *(const float4*)bsrc;
        Bs[buf][bp][(bcol+0)*2 + bo] = f.x;
        Bs[buf][bp][(bcol+1)*2 + bo] = f.y;
        Bs[buf][bp][(bcol+2)*2 + bo] = f.z;
        Bs[buf][bp][(bcol+3)*2 + bo] = f.w;
        if (k0 + 16 < Kd)   // keep the weight stream warm (global_prefetch)
            __builtin_prefetch(bsrc + (size_t)16 * N, 0, 1);
    };

    stage(0, 0);
    wait_async0();
    __syncthreads();

    int buf = 0;
    for (int k0 = 0; k0 < Kd; k0 += 16) {
        if (k0 + 16 < Kd) stage(buf ^ 1, k0 + 16);   // prefetch next chunk (async DMA)

        #pragma unroll
        for (int kk = 0; kk < 16; kk += 4) {
            const int kl = kk + hf * 2;              // even
            const int p  = kl >> 1;
            v2f a; a[0] = As[buf][mRow + ml][kl + 0];
                   a[1] = As[buf][mRow + ml][kl + 1];
            v2f b0, b1, b2, b3;
            b0[0] = Bs[buf][p][( 0 + ml)*2 + 0]; b0[1] = Bs[buf][p][( 0 + ml)*2 + 1];
            b1[0] = Bs[buf][p][(16 + ml)*2 + 0]; b1[1] = Bs[buf][p][(16 + ml)*2 + 1];
            b2[0] = Bs[buf][p][(32 + ml)*2 + 0]; b2[1] = Bs[buf][p][(32 + ml)*2 + 1];
            b3[0] = Bs[buf][p][(48 + ml)*2 + 0]; b3[1] = Bs[buf][p][(48 + ml)*2 + 1];
            acc0 = wmma_f32(a, b0, acc0);
            acc1 = wmma_f32(a, b1, acc1);
            acc2 = wmma_f32(a, b2, acc2);
            acc3 = wmma_f32(a, b3, acc3);
        }
        wait_async0();       // next chunk's DMA landed
        __syncthreads();     // everyone done reading `buf`, writes to buf^1 visible
        buf ^= 1;
    }

    // epilogue: C[mBase+mRow + j + 8*hf][nBase + ns*16 + ml]
    #pragma unroll
    for (int j = 0; j < 8; ++j) {
        const int gm = mBase + mRow + j + 8 * hf;
        float bv0 = bias ? bias[nBase +  0 + ml] : 0.f;
        float bv1 = bias ? bias[nBase + 16 + ml] : 0.f;
        float bv2 = bias ? bias[nBase + 32 + ml] : 0.f;
        float bv3 = bias ? bias[nBase + 48 + ml] : 0.f;
        C[(size_t)gm * N + nBase +  0 + ml] = acc0[j] + bv0;
        C[(size_t)gm * N + nBase + 16 + ml] = acc1[j] + bv1;
        C[(size_t)gm * N + nBase + 32 + ml] = acc2[j] + bv2;
        C[(size_t)gm * N + nBase + 48 + ml] = acc3[j] + bv3;
    }
}

// ---------------------------------------------------------------------------
// Kernel 3: batched biaffine  C[b] = kWb[b](128x1024) @ val_reps[b]^T + bbil
// One block per batch; 8 waves; wave tile 16x128 (8 v8f accumulators).
// A tile staged via async DMA; B tile transposed into padded LDS.
// ---------------------------------------------------------------------------
__global__ __launch_bounds__(256)
void gemm_abt_wmma(const float* __restrict__ A, const float* __restrict__ Bm,
                   const float* __restrict__ bbil, float* __restrict__ C)
{
    __shared__ float As [128][ASTRIDE];
    __shared__ float Bts[16][132];

    const int t    = threadIdx.x;
    const int lane = t & 31;
    const int wave = t >> 5;
    const float* Ab = A  + (size_t)blockIdx.x * K_ * H_;
    const float* Bb = Bm + (size_t)blockIdx.x * K_ * H_;

    v8f acc[8];
    #pragma unroll
    for (int i = 0; i < 8; ++i) acc[i] = (v8f){};

    const int ml   = lane & 15;
    const int hf   = lane >> 4;
    const int mRow = wave * 16;
    const int row  = t >> 1;          // staging: 128 rows x 16 cols
    const int cb   = (t & 1) * 8;

    for (int k0 = 0; k0 < H_; k0 += 16) {
        { // A tile, row-major (async DMA when available)
            const float* sa = Ab + (size_t)row * H_ + k0 + cb;
#if USE_ASYNC_LDS
            async_copy16(&As[row][cb + 0], sa + 0);
            async_copy16(&As[row][cb + 4], sa + 4);
#else
            float4 f0 = *(const float4*)(sa);
            float4 f1 = *(const float4*)(sa + 4);
            As[row][cb+0] = f0.x; As[row][cb+1] = f0.y;
            As[row][cb+2] = f0.z; As[row][cb+3] = f0.w;
            As[row][cb+4] = f1.x; As[row][cb+5] = f1.y;
            As[row][cb+6] = f1.z; As[row][cb+7] = f1.w;
#endif
        }
        { // B tile, transposed into LDS: Bts[k][v] = val_reps[v][k0+k]
            const float* sb = Bb + (size_t)row * H_ + k0 + cb;
            float4 g0 = *(const float4*)(sb);
            float4 g1 = *(const float4*)(sb + 4);
            Bts[cb+0][row] = g0.x; Bts[cb+1][row] = g0.y;
            Bts[cb+2][row] = g0.z; Bts[cb+3][row] = g0.w;
            Bts[cb+4][row] = g1.x; Bts[cb+5][row] = g1.y;
            Bts[cb+6][row] = g1.z; Bts[cb+7][row] = g1.w;
        }
        wait_async0();
        __syncthreads();

        #pragma unroll
        for (int kk = 0; kk < 16; kk += 4) {
            const int kl = kk + hf * 2;
            v2f a; a[0] = As[mRow + ml][kl + 0];
                   a[1] = As[mRow + ml][kl + 1];
            #pragma unroll
            for (int ns = 0; ns < 8; ++ns) {
                v2f b; b[0] = Bts[kl + 0][ns * 16 + ml];
                       b[1] = Bts[kl + 1][ns * 16 + ml];
                acc[ns] = wmma_f32(a, b, acc[ns]);
            }
        }
        __syncthreads();
    }

    const float bb = *bbil;
    float* Cb = C + (size_t)blockIdx.x * K_ * K_;
    #pragma unroll
    for (int ns = 0; ns < 8; ++ns) {
        #pragma unroll
        for (int j = 0; j < 8; ++j) {
            Cb[(mRow + j + 8 * hf) * K_ + ns * 16 + ml] = acc[ns][j] + bb;
        }
    }
}

// ---------------------------------------------------------------------------
// Kernel 4: spatial features + MLP 8->64->32, concat biaffine, MLP 33->16->1.
// One thread per (b,k,v) pair; weights broadcast from LDS; h1 never
// materialized as an array (i-loop rolled, c[33]/f3[16] kept in VGPRs).
// ---------------------------------------------------------------------------
__global__ __launch_bounds__(256)
void fuse_spatial_kernel(const float* __restrict__ bboxes,
                         const int* __restrict__ rowsKey, const int* __restrict__ rowsVal,
                         const float* __restrict__ Ws1, const float* __restrict__ bs1,
                         const float* __restrict__ Ws2, const float* __restrict__ bs2,
                         const float* __restrict__ Wf1, const float* __restrict__ bf1,
                         const float* __restrict__ Wf2, const float* __restrict__ bf2,
                         float* __restrict__ out)
{
    __shared__ float sWs1[8*64], sbs1[64], sWs2[64*32], sbs2[32];
    __shared__ float sWf1[33*16], sbf1[16], sWf2[16], sbf2;
    const int t = threadIdx.x;
    for (int i = t; i < 8*64;  i += 256) sWs1[i] = Ws1[i];
    for (int i = t; i < 64;    i += 256) sbs1[i] = bs1[i];
    for (int i = t; i < 64*32; i += 256) sWs2[i] = Ws2[i];
    for (int i = t; i < 32;    i += 256) sbs2[i] = bs2[i];
    for (int i = t; i < 33*16; i += 256) sWf1[i] = Wf1[i];
    for (int i = t; i < 16;    i += 256) sbf1[i] = bf1[i];
    for (int i = t; i < 16;    i += 256) sWf2[i] = Wf2[i];
    if (t == 0) sbf2 = bf2[0];
    __syncthreads();

    const int g = blockIdx.x * 256 + t;         // 0 .. B*K*K-1
    const int b = g >> 14;
    const int k = (g >> 7) & (K_ - 1);
    const int v = g & (K_ - 1);

    const float* kb = bboxes + (size_t)rowsKey[b*K_ + k] * 4;
    const float* vb = bboxes + (size_t)rowsVal[b*K_ + v] * 4;
    const float kx1=kb[0], ky1=kb[1], kx2=kb[2], ky2=kb[3];
    const float vx1=vb[0], vy1=vb[1], vx2=vb[2], vy2=vb[3];

    const float kcx = (kx1+kx2)*0.5f, kcy = (ky1+ky2)*0.5f;
    const float vcx = (vx1+vx2)*0.5f, vcy = (vy1+vy2)*0.5f;
    const float dx = vcx - kcx, dy = vcy - kcy;
    const float dist = sqrtf(dx*dx + dy*dy + EPS);
    const float ang  = atan2f(dy, dx);
    const float kh = ky2-ky1, vh = vy2-vy1;
    const float h_ov = fmaxf(fminf(ky2, vy2) - fmaxf(ky1, vy1), 0.f);
    const float h_align = h_ov / (fminf(kh, vh) + EPS);
    const float kw = kx2-kx1, vw = vx2-vx1;
    const float v_ov = fmaxf(fminf(kx2, vx2) - fmaxf(kx1, vx1), 0.f);
    const float v_align = v_ov / (fminf(kw, vw) + EPS);
    const float area   = (vh*vw) / ((kh*kw) + EPS);
    const float aspect = (vw / (vh + EPS)) / (kw / (kh + EPS));

    const float sf[8] = {dx, dy, dist, ang, h_align, v_align, area, aspect};

    float c[33];
    c[0] = out[g];                               // biaffine score
    #pragma unroll
    for (int j = 0; j < 32; ++j) c[1+j] = sbs2[j];

    for (int i = 0; i < 64; ++i) {               // rolled: h1_i on the fly
        float h = sbs1[i];
        #pragma unroll
        for (int f = 0; f < 8; ++f) h += sf[f] * sWs1[f*64 + i];
        h = fmaxf(h, 0.f);
        #pragma unroll
        for (int j = 0; j < 32; ++j) c[1+j] += h * sWs2[i*32 + j];
    }

    float f3[16];
    #pragma unroll
    for (int o = 0; o < 16; ++o) f3[o] = sbf1[o];
    #pragma unroll
    for (int m = 0; m < 33; ++m) {
        #pragma unroll
        for (int o = 0; o < 16; ++o) f3[o] += c[m] * sWf1[m*16 + o];
    }
    float score = sbf2;
    #pragma unroll
    for (int o = 0; o < 16; ++o) score += fmaxf(f3[o], 0.f) * sWf2[o];

    out[g] = score;
}

// ---------------------------------------------------------------------------
extern "C" void kernel_launch(void* const* d_in, const int* in_sizes, int n_in,
                              void* d_out, int out_size, void* d_ws, size_t ws_size,
                              hipStream_t stream) {
    const float* seq   = (const float*)d_in[0];
    const float* logit = (const float*)d_in[1];
    const float* bbox  = (const float*)d_in[2];
    const int*   amask = (const int*)  d_in[3];
    const float* Wk    = (const float*)d_in[4];
    const float* bk    = (const float*)d_in[5];
    const float* Wv    = (const float*)d_in[6];
    const float* bv    = (const float*)d_in[7];
    const float* Wbil  = (const float*)d_in[8];
    const float* bbil  = (const float*)d_in[9];
    const float* Ws1   = (const float*)d_in[10];
    const float* bs1   = (const float*)d_in[11];
    const float* Ws2   = (const float*)d_in[12];
    const float* bs2   = (const float*)d_in[13];
    const float* Wf1   = (const float*)d_in[14];
    const float* bf1   = (const float*)d_in[15];
    const float* Wf2   = (const float*)d_in[16];
    const float* bf2   = (const float*)d_in[17];
    float* out = (float*)d_out;

    char* ws = (char*)d_ws;
    int*   rowsKey  = (int*)ws;                       //  512 ints
    int*   rowsVal  = (int*)(ws + 2048);              //  512 ints
    float* key_reps = (float*)(ws + 4096);            //  512*1024 f32
    float* val_reps = key_reps + (size_t)512 * H_;    //  512*1024 f32
    float* kWb      = val_reps + (size_t)512 * H_;    //  512*1024 f32

    // 1) candidate selection (also writes key_idx/val_idx into d_out tail)
    topk_kernel<<<B_ * 2, 256, 0, stream>>>(logit, amask, rowsKey, rowsVal, out);

    // 2-4) projections + bilinear transform (fp32 WMMA, async-LDS double buffer)
    dim3 g1(H_ / 64, 512 / 128);   // (16, 4)
    gemm_rows_wmma<<<g1, 256, 0, stream>>>(seq,      rowsKey, Wk,   bk,      key_reps, 512, H_, H_);
    gemm_rows_wmma<<<g1, 256, 0, stream>>>(seq,      rowsVal, Wv,   bv,      val_reps, 512, H_, H_);
    gemm_rows_wmma<<<g1, 256, 0, stream>>>(key_reps, nullptr, Wbil, nullptr, kWb,      512, H_, H_);

    // 5) biaffine = kWb @ val_reps^T + bbil  -> d_out scores region
    gemm_abt_wmma<<<B_, 256, 0, stream>>>(kWb, val_reps, bbil, out);

    // 6) spatial MLP + fusion head, in-place on scores
    fuse_spatial_kernel<<<B_ * K_ * K_ / 256, 256, 0, stream>>>(
        bbox, rowsKey, rowsVal, Ws1, bs1, Ws2, bs2, Wf1, bf1, Wf2, bf2, out);
}